// TSHAICore_61658550501384
// MI455X (gfx1250) — compile-verified
//
#include <hip/hip_runtime.h>
#include <hip/hip_bf16.h>

typedef __attribute__((ext_vector_type(2))) float v2f;
typedef __attribute__((ext_vector_type(8))) float v8f;

#define NSPLIT 16
#define HALF_LOG2E 0.7213475204444817f   // 0.5 * log2(e)

// ---------------------------------------------------------------------------
// Prep: pack per-particle j-side data and fold beta into charges.
//   jpack[j] = (x_j, y_j, p_amp_j, 1/sigma_j^2)
//   bq[j]    = (beta0*q_j0, beta1*q_j1, beta2*q_j2, beta3*q_j3)
// ---------------------------------------------------------------------------
__global__ void TSHAICore_prep_kernel(const float* __restrict__ pos,
                                      const float* __restrict__ p_amp,
                                      const float* __restrict__ sigma,
                                      const float* __restrict__ charges,
                                      const float* __restrict__ beta,
                                      float4* __restrict__ jpack,
                                      float4* __restrict__ bq, int n) {
    int j = blockIdx.x * blockDim.x + threadIdx.x;
    if (j >= n) return;
    float s = sigma[j];
    float invs2 = 1.0f / (s * s);
    jpack[j] = make_float4(pos[2 * j], pos[2 * j + 1], p_amp[j], invs2);
    bq[j] = make_float4(beta[0] * charges[4 * j + 0],
                        beta[1] * charges[4 * j + 1],
                        beta[2] * charges[4 * j + 2],
                        beta[3] * charges[4 * j + 3]);
}

// ---------------------------------------------------------------------------
// Main pairwise kernel: one wave (32 threads) per (i-tile, j-split).
// w_ij tile (rank-4 beta-weighted charge coupling) via V_WMMA_F32_16X16X4_F32.
// D layout: lane L -> col N = L&15; VGPR d -> row M = d + 8*(L>>4).
// A 16x4 layout: lane L -> row M = L&15; VGPR0/1 -> K = 2*(L>>4), 2*(L>>4)+1.
// B 4x16 layout (mirror): lane L -> col N = L&15; VGPR0/1 -> same K split.
// j-tile operands are software-pipelined (double-buffered in registers) so
// the L2 load latency overlaps the previous tile's 8-row VALU body instead of
// stalling in front of the WMMA.
// ---------------------------------------------------------------------------
__global__ __launch_bounds__(32)
void TSHAICore_pair_kernel(const float* __restrict__ pos,
                           const float* __restrict__ charges,
                           const float4* __restrict__ jpack,
                           const float4* __restrict__ bq,
                           float* __restrict__ partials, int n) {
    const int lane = threadIdx.x;        // 0..31
    const int hi   = lane >> 4;          // half of wave (K split / M split)
    const int lo   = lane & 15;          // column within tile
    const int ti   = blockIdx.x;         // i-tile index
    const int js   = blockIdx.y;         // j-split index
    const int i0   = ti * 16;
    const int ntiles = n >> 4;

    // A fragment: charges of this lane's i-row, K pair selected by half.
    v2f afrag;
    {
        const float* cp = charges + 4 * (i0 + lo) + 2 * hi;
        afrag.x = cp[0];
        afrag.y = cp[1];
    }

    // i-side positions for the 8 rows this lane accumulates (M = hi*8 + v).
    float xi[8], yi[8];
#pragma unroll
    for (int v = 0; v < 8; ++v) {
        int ii = i0 + hi * 8 + v;
        xi[v] = pos[2 * ii];
        yi[v] = pos[2 * ii + 1];
    }

    float ptot[8], p1x[8], p1y[8], p2x[8], p2y[8];
#pragma unroll
    for (int v = 0; v < 8; ++v) {
        ptot[v] = 0.f; p1x[v] = 0.f; p1y[v] = 0.f; p2x[v] = 0.f; p2y[v] = 0.f;
    }

    const v8f czero = {0.f, 0.f, 0.f, 0.f, 0.f, 0.f, 0.f, 0.f};

    // ---- software pipeline: preload first j-tile operands ----
    float4 jd;
    v2f    bfrag;
    if (js < ntiles) {
        const int j = js * 16 + lo;
        jd = jpack[j];
        const float* bp = reinterpret_cast<const float*>(bq + j) + 2 * hi;
        bfrag.x = bp[0];
        bfrag.y = bp[1];
    }

    for (int jt = js; jt < ntiles; jt += NSPLIT) {
        // issue next tile's loads before touching this tile's data
        const int jtn = jt + NSPLIT;
        float4 jd_n = jd;
        v2f    b_n  = bfrag;
        if (jtn < ntiles) {
            const int jn = jtn * 16 + lo;
            jd_n = jpack[jn];
            const float* bp = reinterpret_cast<const float*>(bq + jn) + 2 * hi;
            b_n.x = bp[0];
            b_n.y = bp[1];
        }

        const float e2 = jd.w * (-HALF_LOG2E);

        // w[16x16] = A[16x4] x B[4x16] : one wmma, 2048 flops
        v8f w = __builtin_amdgcn_wmma_f32_16x16x4_f32(
            false, afrag, false, bfrag, (short)0, czero, false, false);

#pragma unroll
        for (int v = 0; v < 8; ++v) {
            float dx = xi[v] - jd.x;
            float dy = yi[v] - jd.y;
            float d2 = __builtin_fmaf(dy, dy, dx * dx);
            float e  = __builtin_amdgcn_exp2f(d2 * e2);   // v_exp_f32
            float p  = jd.z * e;
            ptot[v] += p;
            float c  = p * jd.w;                          // p / sigma_j^2
            p1x[v] = __builtin_fmaf(dx, c, p1x[v]);
            p1y[v] = __builtin_fmaf(dy, c, p1y[v]);
            float wc = w[v] * c;
            p2x[v] = __builtin_fmaf(dx, wc, p2x[v]);
            p2y[v] = __builtin_fmaf(dy, wc, p2y[v]);
        }

        jd    = jd_n;
        bfrag = b_n;
    }

    // Butterfly sum across the 16 lanes of each half (xor of bits 0..3 stays
    // inside the half on wave32).
#pragma unroll
    for (int v = 0; v < 8; ++v) {
#pragma unroll
        for (int m = 1; m < 16; m <<= 1) {
            ptot[v] += __shfl_xor(ptot[v], m, 32);
            p1x[v]  += __shfl_xor(p1x[v],  m, 32);
            p1y[v]  += __shfl_xor(p1y[v],  m, 32);
            p2x[v]  += __shfl_xor(p2x[v],  m, 32);
            p2y[v]  += __shfl_xor(p2y[v],  m, 32);
        }
    }

    if (lo == 0) {   // lane 0 writes rows 0..7, lane 16 writes rows 8..15
#pragma unroll
        for (int v = 0; v < 8; ++v) {
            int M = hi * 8 + v;
            float* dst = partials +
                (((size_t)js * (size_t)ntiles + (size_t)ti) * 16 + M) * 5;
            dst[0] = ptot[v];
            dst[1] = p1x[v];
            dst[2] = p1y[v];
            dst[3] = p2x[v];
            dst[4] = p2y[v];
        }
    }
}

// ---------------------------------------------------------------------------
// Combine split-K partials; emit f_unified [N,2] then p_total [N].
// f_unified = alpha*P1/p_total + P2  (sign algebra folded)
// ---------------------------------------------------------------------------
__global__ void TSHAICore_final_kernel(const float* __restrict__ partials,
                                       const float* __restrict__ alpha_p,
                                       float* __restrict__ out, int n) {
    int i = blockIdx.x * blockDim.x + threadIdx.x;
    if (i >= n) return;
    const float alpha = alpha_p[0];
    const int ntiles = n >> 4;
    const int ti = i >> 4, M = i & 15;
    float ptot = 0.f, p1x = 0.f, p1y = 0.f, p2x = 0.f, p2y = 0.f;
    for (int s = 0; s < NSPLIT; ++s) {
        const float* src = partials +
            (((size_t)s * (size_t)ntiles + (size_t)ti) * 16 + M) * 5;
        ptot += src[0]; p1x += src[1]; p1y += src[2]; p2x += src[3]; p2y += src[4];
    }
    ptot += 1e-7f;
    float inv = alpha / ptot;
    out[2 * i + 0] = p1x * inv + p2x;
    out[2 * i + 1] = p1y * inv + p2y;
    out[2 * n + i] = ptot;
}

extern "C" void kernel_launch(void* const* d_in, const int* in_sizes, int n_in,
                              void* d_out, int out_size, void* d_ws, size_t ws_size,
                              hipStream_t stream) {
    const float* pos     = (const float*)d_in[0];
    const float* p_amp   = (const float*)d_in[1];
    const float* sigma   = (const float*)d_in[2];
    const float* charges = (const float*)d_in[3];
    const float* alpha   = (const float*)d_in[4];
    const float* beta    = (const float*)d_in[5];
    const int n = in_sizes[1];               // 8192 particles

    char* ws = (char*)d_ws;
    float4* jpack    = (float4*)ws;                          // n*16 bytes
    float4* bq       = (float4*)(ws + (size_t)n * 16);       // n*16 bytes
    float*  partials = (float*) (ws + (size_t)n * 32);       // NSPLIT*n*5 floats

    TSHAICore_prep_kernel<<<(n + 255) / 256, 256, 0, stream>>>(
        pos, p_amp, sigma, charges, beta, jpack, bq, n);

    dim3 grid(n / 16, NSPLIT);
    TSHAICore_pair_kernel<<<grid, 32, 0, stream>>>(
        pos, charges, jpack, bq, partials, n);

    TSHAICore_final_kernel<<<(n + 255) / 256, 256, 0, stream>>>(
        partials, alpha, (float*)d_out, n);
}